// Model_90280212562327
// MI455X (gfx1250) — compile-verified
//
#include <hip/hip_runtime.h>

typedef __attribute__((ext_vector_type(2))) float v2f;
typedef __attribute__((ext_vector_type(8))) float v8f;

#define D_MODEL 128
#define NCH 64
#define NH 8
#define DH 16

// ---------------- CDNA5 async global->LDS helpers ----------------
// Copies 16B per lane from a per-lane global address into LDS (ASYNCcnt-tracked).
__device__ inline void async_copy_b128(const float* gaddr, float* lds_ptr) {
  unsigned lds = (unsigned)(uintptr_t)lds_ptr;  // low 32 bits of generic LDS addr = LDS offset
  asm volatile("global_load_async_to_lds_b128 %0, %1, off"
               :: "v"(lds), "v"(gaddr)
               : "memory");
}
__device__ inline void async_wait0() {
  asm volatile("s_wait_asynccnt 0x0" ::: "memory");
}

// ---------------- WMMA helpers (V_WMMA_F32_16X16X4_F32) ----------------
// A 16x4 fragment: lanes0-15 -> M=lane, v0=K0, v1=K1 ; lanes16-31 -> v0=K2, v1=K3
// B 4x16 fragment: mirrored (lanes = N, v0=K0/K2, v1=K1/K3)
// C/D 16x16: VGPR r, lanes0-15 M=r N=lane ; lanes16-31 M=r+8

__device__ inline v8f wmma4(v2f a, v2f b, v8f c) {
  return __builtin_amdgcn_wmma_f32_16x16x4_f32(false, a, false, b, (short)0, c,
                                               false, false);
}

// C16x16 += A(16xK, LDS row-major lda) * B(KxN via weight W row-major NxK; Wg pre-offset to row n0)
__device__ inline v8f tile_AW(const float* Alds, int lda, const float* Wg, int ldw,
                              int K, int lane, v8f acc) {
  int m = lane & 15, hi = lane >> 4;
  for (int k = 0; k < K; k += 4) {
    int ka = k + 2 * hi;
    v2f a, b;
    a.x = Alds[m * lda + ka];       a.y = Alds[m * lda + ka + 1];
    b.x = Wg[m * (size_t)ldw + ka]; b.y = Wg[m * (size_t)ldw + ka + 1];
    acc = wmma4(a, b, acc);
  }
  return acc;
}

// C16x16 += A(16xK, LDS row-major lda) * B(KxN, LDS row-major ldb; Blds pre-offset to col n0)
__device__ inline v8f tile_AB(const float* Alds, int lda, const float* Blds, int ldb,
                              int K, int lane, v8f acc) {
  int m = lane & 15, hi = lane >> 4;
  for (int k = 0; k < K; k += 4) {
    int ka = k + 2 * hi;
    v2f a, b;
    a.x = Alds[m * lda + ka];  a.y = Alds[m * lda + ka + 1];
    b.x = Blds[ka * ldb + m];  b.y = Blds[(ka + 1) * ldb + m];
    acc = wmma4(a, b, acc);
  }
  return acc;
}

// Gram tile: C = Xc^T[m0 rows] * Xc[n0 cols], Xc is Lp x 64 row-major in LDS
__device__ inline v8f tile_gram(const float* Xc, int m0, int n0, int Kp, int lane, v8f acc) {
  int m = lane & 15, hi = lane >> 4;
  for (int k = 0; k < Kp; k += 4) {
    int ka = k + 2 * hi;
    v2f a, b;
    a.x = Xc[ka * NCH + m0 + m]; a.y = Xc[(ka + 1) * NCH + m0 + m];
    b.x = Xc[ka * NCH + n0 + m]; b.y = Xc[(ka + 1) * NCH + n0 + m];
    acc = wmma4(a, b, acc);
  }
  return acc;
}

// ---------------- fused per-window encoder ----------------
// One block per (b,s): stats -> corr -> top-k binarize -> GIN x2 -> readout pool -> G(+posenc)
__global__ void __launch_bounds__(256) encode_kernel(
    const float* __restrict__ x, int T, int S, int win, int step,
    const float* __restrict__ nn_g, const float* __restrict__ nn_b,
    const float* __restrict__ eps0, const float* __restrict__ W1_0,
    const float* __restrict__ b1_0, const float* __restrict__ W2_0,
    const float* __restrict__ b2_0,
    const float* __restrict__ eps1, const float* __restrict__ W1_1,
    const float* __restrict__ b1_1, const float* __restrict__ W2_1,
    const float* __restrict__ b2_1,
    const float* __restrict__ rW, const float* __restrict__ rb,
    const float* __restrict__ ru,
    float* __restrict__ G) {
  extern __shared__ float smem[];
  float* Xc   = smem;            // 64x64  (Lp x NCH, zero padded)
  float* Adj  = smem + 4096;     // 64x64  (cov -> corr -> adjacency)
  float* Hb   = smem + 8192;     // 64x128
  float* Mb   = smem + 16384;    // 64x128
  float* st   = smem + 24576;
  float* meanv = st;             // 64
  float* sdv   = st + 64;        // 64
  float* rstd  = st + 128;       // 64
  float* X3    = st + 192;       // 64x3
  float* agg3  = st + 384;       // 64x3 (later: node scores)
  int*   cnti  = (int*)(st + 576);

  const int tid  = threadIdx.x;
  const int lane = tid & 31;
  const int wv   = tid >> 5;
  const int bs = blockIdx.x;
  const int b = bs / S, s = bs % S;
  const int t0 = s * step;
  const int L = win;
  const int Lp = (L + 3) & ~3;

  // warm L2/WGP$ for the weight matrices consumed later (global_prefetch_b8)
  __builtin_prefetch(W2_0 + (size_t)tid * 64, 0, 0);
  __builtin_prefetch(W1_1 + (size_t)tid * 64, 0, 0);
  __builtin_prefetch(W2_1 + (size_t)tid * 64, 0, 0);
  __builtin_prefetch(rW   + (size_t)tid * 64, 0, 0);

  // ---- async load window into LDS (contiguous L*64 floats), zero pad to Lp ----
  {
    const float* xw = x + ((size_t)b * T + t0) * NCH;
    for (int i = tid * 4; i < L * NCH; i += 1024)
      async_copy_b128(xw + i, Xc + i);
    for (int i = L * NCH + tid; i < Lp * NCH; i += 256) Xc[i] = 0.0f;
    async_wait0();
  }
  __syncthreads();
  // ---- per-channel mean / std (ddof=1) ----
  if (tid < NCH) {
    float m = 0.0f;
    for (int t = 0; t < L; ++t) m += Xc[t * NCH + tid];
    m /= (float)L;
    float v = 0.0f;
    for (int t = 0; t < L; ++t) { float d = Xc[t * NCH + tid] - m; v += d * d; }
    meanv[tid] = m;
    sdv[tid] = fmaxf(sqrtf(v / (float)(L - 1)), 1e-8f);
  }
  __syncthreads();
  for (int i = tid; i < L * NCH; i += 256) Xc[i] -= meanv[i & 63];
  __syncthreads();

  // ---- cov = Xc^T Xc / (L-1+1e-8) via WMMA (16 tiles, 8 waves x 2) ----
  {
    const float inv = 1.0f / ((float)(L - 1) + 1e-8f);
    for (int t = wv; t < 16; t += 8) {
      int tm = t & 3, tn = t >> 2;
      v8f acc = {};
      acc = tile_gram(Xc, tm * 16, tn * 16, Lp, lane, acc);
      int n = lane & 15, hi = lane >> 4;
#pragma unroll
      for (int r = 0; r < 8; ++r)
        Adj[(tm * 16 + r + 8 * hi) * NCH + tn * 16 + n] = acc[r] * inv;
    }
  }
  __syncthreads();
  // ---- corr ----
  if (tid < NCH) rstd[tid] = rsqrtf(fmaxf(Adj[tid * NCH + tid], 1e-8f));
  __syncthreads();
  for (int i = tid; i < NCH * NCH; i += 256) {
    int m = i >> 6, n = i & 63;
    float c = Adj[i] * rstd[m] * rstd[n];
    if (!(fabsf(c) < 1e38f)) c = 0.0f;  // nan_to_num
    Adj[i] = (m == n) ? 1.0f : c;
  }
  __syncthreads();
  // ---- top-k threshold (k-th largest |corr| of 2016 upper-tri) by bisection ----
  const int ktop = 605;  // round(0.3 * 64*63/2)
  float lo = 0.0f, hi2 = 1.0001f;
  for (int it = 0; it < 28; ++it) {
    float mid = 0.5f * (lo + hi2);
    if (tid == 0) cnti[0] = 0;
    __syncthreads();
    int c = 0;
    for (int i = tid; i < NCH * NCH; i += 256) {
      int m = i >> 6, n = i & 63;
      if (m < n && fabsf(Adj[i]) >= mid) ++c;
    }
    atomicAdd(cnti, c);
    __syncthreads();
    int tot = cnti[0];
    __syncthreads();
    if (tot >= ktop) lo = mid; else hi2 = mid;
  }
  // ---- binarize (symmetric) + identity ----
  for (int i = tid; i < NCH * NCH; i += 256) {
    int m = i >> 6, n = i & 63;
    Adj[i] = (m == n) ? 1.0f : ((fabsf(Adj[i]) >= lo) ? 1.0f : 0.0f);
  }
  __syncthreads();
  // ---- node features + layernorm over 3 ----
  if (tid < NCH) {
    float deg = 0.0f;
    for (int n = 0; n < NCH; ++n) deg += Adj[tid * NCH + n];
    deg -= 1.0f;
    float f0 = meanv[tid], f1 = sdv[tid], f2 = deg;
    float mu = (f0 + f1 + f2) * (1.0f / 3.0f);
    float va = ((f0 - mu) * (f0 - mu) + (f1 - mu) * (f1 - mu) +
                (f2 - mu) * (f2 - mu)) * (1.0f / 3.0f);
    float inv = rsqrtf(va + 1e-5f);
    X3[tid * 3 + 0] = (f0 - mu) * inv * nn_g[0] + nn_b[0];
    X3[tid * 3 + 1] = (f1 - mu) * inv * nn_g[1] + nn_b[1];
    X3[tid * 3 + 2] = (f2 - mu) * inv * nn_g[2] + nn_b[2];
  }
  __syncthreads();
  // ---- GIN layer 0 (fin=3): agg = (1+eps)X + A X ----
  float e0 = 1.0f + eps0[0];
  for (int i = tid; i < NCH * 3; i += 256) {
    int m = i / 3, f = i - m * 3;
    float acc = e0 * X3[i];
    for (int n = 0; n < NCH; ++n) acc += Adj[m * NCH + n] * X3[n * 3 + f];
    agg3[i] = acc;
  }
  __syncthreads();
  // Mb = relu(agg3 @ W1_0^T + b1_0)  (K=3, scalar)
  for (int i = tid; i < NCH * D_MODEL; i += 256) {
    int m = i >> 7, j = i & 127;
    float acc = b1_0[j] + agg3[m * 3 + 0] * W1_0[j * 3 + 0] +
                agg3[m * 3 + 1] * W1_0[j * 3 + 1] +
                agg3[m * 3 + 2] * W1_0[j * 3 + 2];
    Mb[i] = fmaxf(acc, 0.0f);
  }
  __syncthreads();
  // Hb = Mb @ W2_0^T + b2_0   (WMMA, 32 tiles, 8 waves x 4)
  for (int t = wv; t < 32; t += 8) {
    int tm = t & 3, tn = t >> 2;
    v8f acc = {};
    acc = tile_AW(&Mb[tm * 16 * D_MODEL], D_MODEL, &W2_0[(size_t)(tn * 16) * D_MODEL],
                  D_MODEL, D_MODEL, lane, acc);
    int n = lane & 15, hi = lane >> 4;
#pragma unroll
    for (int r = 0; r < 8; ++r)
      Hb[(tm * 16 + r + 8 * hi) * D_MODEL + tn * 16 + n] = acc[r] + b2_0[tn * 16 + n];
  }
  __syncthreads();
  // ---- GIN layer 1: Mb(agg) = (1+eps1)Hb + Adj @ Hb (WMMA K=64) ----
  float e1 = 1.0f + eps1[0];
  for (int t = wv; t < 32; t += 8) {
    int tm = t & 3, tn = t >> 2;
    v8f acc = {};
    acc = tile_AB(&Adj[tm * 16 * NCH], NCH, &Hb[tn * 16], D_MODEL, NCH, lane, acc);
    int n = lane & 15, hi = lane >> 4;
#pragma unroll
    for (int r = 0; r < 8; ++r) {
      int row = tm * 16 + r + 8 * hi, col = tn * 16 + n;
      Mb[row * D_MODEL + col] = acc[r] + e1 * Hb[row * D_MODEL + col];
    }
  }
  __syncthreads();
  // Hb = relu(Mb @ W1_1^T + b1_1)
  for (int t = wv; t < 32; t += 8) {
    int tm = t & 3, tn = t >> 2;
    v8f acc = {};
    acc = tile_AW(&Mb[tm * 16 * D_MODEL], D_MODEL, &W1_1[(size_t)(tn * 16) * D_MODEL],
                  D_MODEL, D_MODEL, lane, acc);
    int n = lane & 15, hi = lane >> 4;
#pragma unroll
    for (int r = 0; r < 8; ++r)
      Hb[(tm * 16 + r + 8 * hi) * D_MODEL + tn * 16 + n] =
          fmaxf(acc[r] + b1_1[tn * 16 + n], 0.0f);
  }
  __syncthreads();
  // Mb = Hb @ W2_1^T + b2_1   (final H lives in Mb)
  for (int t = wv; t < 32; t += 8) {
    int tm = t & 3, tn = t >> 2;
    v8f acc = {};
    acc = tile_AW(&Hb[tm * 16 * D_MODEL], D_MODEL, &W2_1[(size_t)(tn * 16) * D_MODEL],
                  D_MODEL, D_MODEL, lane, acc);
    int n = lane & 15, hi = lane >> 4;
#pragma unroll
    for (int r = 0; r < 8; ++r)
      Mb[(tm * 16 + r + 8 * hi) * D_MODEL + tn * 16 + n] = acc[r] + b2_1[tn * 16 + n];
  }
  __syncthreads();
  // ---- readout: Hb = tanh(Mb @ rW^T + rb) ----
  for (int t = wv; t < 32; t += 8) {
    int tm = t & 3, tn = t >> 2;
    v8f acc = {};
    acc = tile_AW(&Mb[tm * 16 * D_MODEL], D_MODEL, &rW[(size_t)(tn * 16) * D_MODEL],
                  D_MODEL, D_MODEL, lane, acc);
    int n = lane & 15, hi = lane >> 4;
#pragma unroll
    for (int r = 0; r < 8; ++r)
      Hb[(tm * 16 + r + 8 * hi) * D_MODEL + tn * 16 + n] =
          tanhf(acc[r] + rb[tn * 16 + n]);
  }
  __syncthreads();
  if (tid < NCH) {
    float sc = 0.0f;
    for (int j = 0; j < D_MODEL; ++j) sc += Hb[tid * D_MODEL + j] * ru[j];
    agg3[tid] = sc;
  }
  __syncthreads();
  // ---- softmax over 64 nodes, pool, posenc, write G ----
  if (tid < D_MODEL) {
    float mx = -1e30f;
    for (int m = 0; m < NCH; ++m) mx = fmaxf(mx, agg3[m]);
    float sum = 0.0f;
    for (int m = 0; m < NCH; ++m) sum += __expf(agg3[m] - mx);
    float acc = 0.0f;
    for (int m = 0; m < NCH; ++m)
      acc += __expf(agg3[m] - mx) * Mb[m * D_MODEL + tid];
    acc /= sum;
    float div = __expf(-(float)(tid & ~1) * (logf(10000.0f) / (float)D_MODEL));
    float pe = (tid & 1) ? cosf((float)s * div) : sinf((float)s * div);
    G[((size_t)b * S + s) * D_MODEL + tid] = acc + pe;
  }
}

// ---------------- generic WMMA GEMM: C = act(A @ W^T + bias) (+res) ----------------
// A: M x K row-major. W: N x K row-major. 64x64 output tile per block, K in 128 chunks.
__global__ void __launch_bounds__(256) gemm_kernel(
    const float* __restrict__ A, const float* __restrict__ W,
    const float* __restrict__ bias, const float* __restrict__ res,
    float* __restrict__ C, int M, int N, int K, int act) {
  extern __shared__ float As[];  // 64 x 128
  const int tid = threadIdx.x, lane = tid & 31, wv = tid >> 5;
  const int m0 = blockIdx.x * 64;
  const int n0 = blockIdx.y * 64;
  // prefetch this block's weight panel (64 rows x K)
  __builtin_prefetch(W + (size_t)n0 * K + (size_t)tid * 64, 0, 0);
  if (K > 128)
    __builtin_prefetch(W + (size_t)n0 * K + 16384 + (size_t)tid * 64, 0, 0);
  v8f acc0 = {}, acc1 = {};
  for (int kc = 0; kc < K; kc += 128) {
    // async stage A tile (64 x 128) into LDS; OOB rows clamped (their C rows are masked)
    for (int i = tid * 4; i < 64 * 128; i += 1024) {
      int r = i >> 7, k = i & 127;
      int gr = m0 + r;
      if (gr >= M) gr = M - 1;
      async_copy_b128(A + (size_t)gr * K + kc + k, As + i);
    }
    async_wait0();
    __syncthreads();
    {
      int t = wv, tm = t & 3, tn = t >> 2;
      acc0 = tile_AW(&As[tm * 16 * 128], 128,
                     &W[(size_t)(n0 + tn * 16) * K + kc], K, 128, lane, acc0);
      t = wv + 8; tm = t & 3; tn = t >> 2;
      acc1 = tile_AW(&As[tm * 16 * 128], 128,
                     &W[(size_t)(n0 + tn * 16) * K + kc], K, 128, lane, acc1);
    }
    __syncthreads();
  }
  int n = lane & 15, hi = lane >> 4;
  for (int p = 0; p < 2; ++p) {
    int t = wv + 8 * p, tm = t & 3, tn = t >> 2;
    v8f acc = p ? acc1 : acc0;
#pragma unroll
    for (int r = 0; r < 8; ++r) {
      int row = m0 + tm * 16 + r + 8 * hi;
      int col = n0 + tn * 16 + n;
      if (row < M) {
        float v = acc[r] + bias[col];
        if (act == 1) v = fmaxf(v, 0.0f);
        else if (act == 2) v = 0.5f * v * (1.0f + erff(v * 0.70710678f));  // exact gelu
        else if (act == 3) v = tanhf(v);
        if (res) v += res[(size_t)row * N + col];
        C[(size_t)row * N + col] = v;
      }
    }
  }
}

// ---------------- attention: per (b,h), K/V head staged in LDS ----------------
__global__ void __launch_bounds__(256) attn_kernel(const float* __restrict__ QKV,
                                                   float* __restrict__ O, int S) {
  extern __shared__ float smem[];
  float* Kh = smem;
  float* Vh = smem + S * DH;
  const int tid = threadIdx.x;
  const int b = blockIdx.x / NH, h = blockIdx.x % NH;
  const size_t base = (size_t)b * S * 384;
  for (int i = tid; i < S * DH; i += 256) {
    int sr = i >> 4, j = i & 15;
    Kh[i] = QKV[base + (size_t)sr * 384 + 128 + h * DH + j];
    Vh[i] = QKV[base + (size_t)sr * 384 + 256 + h * DH + j];
  }
  __syncthreads();
  for (int q = tid; q < S; q += 256) {
    float qv[DH];
    for (int j = 0; j < DH; ++j) qv[j] = QKV[base + (size_t)q * 384 + h * DH + j];
    float mx = -1e30f;
    for (int ks = 0; ks < S; ++ks) {
      float d = 0.0f;
      for (int j = 0; j < DH; ++j) d += qv[j] * Kh[ks * DH + j];
      mx = fmaxf(mx, d * 0.25f);
    }
    float sum = 0.0f, o[DH];
    for (int j = 0; j < DH; ++j) o[j] = 0.0f;
    for (int ks = 0; ks < S; ++ks) {
      float d = 0.0f;
      for (int j = 0; j < DH; ++j) d += qv[j] * Kh[ks * DH + j];
      float p = __expf(d * 0.25f - mx);
      sum += p;
      for (int j = 0; j < DH; ++j) o[j] += p * Vh[ks * DH + j];
    }
    float inv = 1.0f / sum;
    for (int j = 0; j < DH; ++j)
      O[((size_t)b * S + q) * D_MODEL + h * DH + j] = o[j] * inv;
  }
}

// ---------------- layernorm over 128 ----------------
__global__ void ln_kernel(const float* __restrict__ X, float* __restrict__ Y,
                          const float* __restrict__ g, const float* __restrict__ bta) {
  __shared__ float red[128];
  int row = blockIdx.x, j = threadIdx.x;
  float v = X[(size_t)row * D_MODEL + j];
  red[j] = v; __syncthreads();
  for (int off = 64; off > 0; off >>= 1) { if (j < off) red[j] += red[j + off]; __syncthreads(); }
  float mu = red[0] * (1.0f / D_MODEL);
  __syncthreads();
  float d = v - mu;
  red[j] = d * d; __syncthreads();
  for (int off = 64; off > 0; off >>= 1) { if (j < off) red[j] += red[j + off]; __syncthreads(); }
  float var = red[0] * (1.0f / D_MODEL);
  Y[(size_t)row * D_MODEL + j] = d * rsqrtf(var + 1e-5f) * g[j] + bta[j];
}

// ---------------- sequence attention pool (axis=1), writes z columns ----------------
__global__ void __launch_bounds__(256) pool_kernel(
    const float* __restrict__ G, const float* __restrict__ W,
    const float* __restrict__ bias, const float* __restrict__ v,
    float* __restrict__ z, int S, int zld, int zoff) {
  extern __shared__ float sc[];  // S scores
  int b = blockIdx.x, tid = threadIdx.x;
  const float* Gb = G + (size_t)b * S * D_MODEL;
  for (int sr = tid; sr < S; sr += 256) {
    float score = 0.0f;
    const float* gr = Gb + (size_t)sr * D_MODEL;
    for (int j = 0; j < D_MODEL; ++j) {
      float a = bias[j];
      const float* wj = W + (size_t)j * D_MODEL;
      for (int k = 0; k < D_MODEL; ++k) a += gr[k] * wj[k];
      score += tanhf(a) * v[j];
    }
    sc[sr] = score;
  }
  __syncthreads();
  if (tid < D_MODEL) {
    float mx = -1e30f;
    for (int i = 0; i < S; ++i) mx = fmaxf(mx, sc[i]);
    float sum = 0.0f;
    for (int i = 0; i < S; ++i) sum += __expf(sc[i] - mx);
    float acc = 0.0f;
    for (int i = 0; i < S; ++i) acc += __expf(sc[i] - mx) * Gb[(size_t)i * D_MODEL + tid];
    z[(size_t)b * zld + zoff + tid] = acc / sum;
  }
}

// ---------------- classifier ----------------
__global__ void cls_kernel(const float* __restrict__ z, const float* __restrict__ W,
                           const float* __restrict__ bias, float* __restrict__ out, int B) {
  int b = threadIdx.x;
  if (b < B) {
    float a = bias[0];
    for (int j = 0; j < 2 * D_MODEL; ++j) a += z[b * 2 * D_MODEL + j] * W[j];
    out[b] = 1.0f / (1.0f + __expf(-a));
  }
}

// ---------------- host orchestration ----------------
extern "C" void kernel_launch(void* const* d_in, const int* in_sizes, int n_in,
                              void* d_out, int out_size, void* d_ws, size_t ws_size,
                              hipStream_t stream) {
  (void)in_sizes; (void)n_in; (void)out_size; (void)ws_size;
  // input order: x_enc, then params as a sorted-key pytree:
  // cls{W,b}, gin[0]{W1,W2,b1,b2,eps}, gin[1]{...}, node_norm{b,g},
  // pool{W,b,v}, readout{W,b,u}, tx[0]{W1,W2,Win,Wout,b1,b2,bin,bout,ln1_b,ln1_g,ln2_b,ln2_g}, tx[1]{...}
  const float* x    = (const float*)d_in[0];
  const float* clsW = (const float*)d_in[1];
  const float* clsB = (const float*)d_in[2];
  const float* g0W1 = (const float*)d_in[3];
  const float* g0W2 = (const float*)d_in[4];
  const float* g0b1 = (const float*)d_in[5];
  const float* g0b2 = (const float*)d_in[6];
  const float* g0e  = (const float*)d_in[7];
  const float* g1W1 = (const float*)d_in[8];
  const float* g1W2 = (const float*)d_in[9];
  const float* g1b1 = (const float*)d_in[10];
  const float* g1b2 = (const float*)d_in[11];
  const float* g1e  = (const float*)d_in[12];
  const float* nnb  = (const float*)d_in[13];
  const float* nng  = (const float*)d_in[14];
  const float* poW  = (const float*)d_in[15];
  const float* poB  = (const float*)d_in[16];
  const float* poV  = (const float*)d_in[17];
  const float* rW   = (const float*)d_in[18];
  const float* rb   = (const float*)d_in[19];
  const float* ru   = (const float*)d_in[20];
  const float* tx[2][12];
  for (int l = 0; l < 2; ++l)
    for (int k = 0; k < 12; ++k) tx[l][k] = (const float*)d_in[21 + 12 * l + k];
  // per-layer: 0=W1 1=W2 2=Win 3=Wout 4=b1 5=b2 6=bin 7=bout 8=ln1_b 9=ln1_g 10=ln2_b 11=ln2_g

  const int B = 16, T = 512;
  const size_t SMAX = 242;
  float* ws  = (float*)d_ws;
  float* G   = ws;                        // B*SMAX*128
  float* QKV = G + B * SMAX * 128;        // B*SMAX*384
  float* T1  = QKV + B * SMAX * 384;      // B*SMAX*128
  float* T2  = T1 + B * SMAX * 128;       // B*SMAX*128
  float* FH  = T2 + B * SMAX * 128;       // B*SMAX*256
  float* Z2  = FH + B * SMAX * 256;       // B*256

  const int wins[2] = {30, 60};
  const size_t enc_lds = (24576 + 640) * sizeof(float);
  for (int sc2 = 0; sc2 < 2; ++sc2) {
    int win = wins[sc2], step = 2;
    int S = (T - win) / step + 1;
    int MR = B * S;
    encode_kernel<<<dim3(MR), dim3(256), enc_lds, stream>>>(
        x, T, S, win, step, nng, nnb,
        g0e, g0W1, g0b1, g0W2, g0b2,
        g1e, g1W1, g1b1, g1W2, g1b2,
        rW, rb, ru, G);
    int gx = (MR + 63) / 64;
    size_t glds = 64 * 128 * sizeof(float);
    for (int l = 0; l < 2; ++l) {
      // QKV = G @ Win^T + bin
      gemm_kernel<<<dim3(gx, 6), 256, glds, stream>>>(G, tx[l][2], tx[l][6], nullptr,
                                                      QKV, MR, 384, 128, 0);
      // T1 = attention(QKV)
      attn_kernel<<<dim3(B * NH), 256, (size_t)S * DH * 2 * sizeof(float), stream>>>(QKV, T1, S);
      // T2 = T1 @ Wout^T + bout + G (residual)
      gemm_kernel<<<dim3(gx, 2), 256, glds, stream>>>(T1, tx[l][3], tx[l][7], G,
                                                      T2, MR, 128, 128, 0);
      // T1 = LN1(T2)
      ln_kernel<<<dim3(MR), 128, 0, stream>>>(T2, T1, tx[l][9], tx[l][8]);
      // FH = gelu(T1 @ W1^T + b1)
      gemm_kernel<<<dim3(gx, 4), 256, glds, stream>>>(T1, tx[l][0], tx[l][4], nullptr,
                                                      FH, MR, 256, 128, 2);
      // T2 = FH @ W2^T + b2 + T1
      gemm_kernel<<<dim3(gx, 2), 256, glds, stream>>>(FH, tx[l][1], tx[l][5], T1,
                                                      T2, MR, 128, 256, 0);
      // G = LN2(T2)
      ln_kernel<<<dim3(MR), 128, 0, stream>>>(T2, G, tx[l][11], tx[l][10]);
    }
    pool_kernel<<<dim3(B), 256, (size_t)S * sizeof(float), stream>>>(
        G, poW, poB, poV, Z2, S, 256, sc2 * 128);
  }
  cls_kernel<<<dim3(1), 32, 0, stream>>>(Z2, clsW, clsB, (float*)d_out, B);
}